// roiAlign_47639777247673
// MI455X (gfx1250) — compile-verified
//
#include <hip/hip_runtime.h>
#include <hip/hip_bf16.h>

typedef __attribute__((ext_vector_type(2))) float v2f;
typedef __attribute__((ext_vector_type(8))) float v8f;

#define SCALE_F 0.0625f
#define C_TOT 1024
#define H_IN 50
#define W_IN 50
#define PH_N 7
#define PW_N 7

__global__ __launch_bounds__(256) void roialign_wmma_f32(
    const float* __restrict__ data, const float* __restrict__ rois,
    float* __restrict__ out)
{
  __shared__ __align__(16) float BxW[16][16];   // [x_local][pw]  (K x N for stage 1)
  __shared__ __align__(16) float AyW[16][16];   // [ph][y_local]  (M x K for stage 2)
  __shared__ __align__(16) float UT[8][16][18]; // per-wave U^T: [pw][y_local], pitch 18
  __shared__ int sp[3];                         // b, y0, x0

  const int n    = blockIdx.y;
  const int grp  = blockIdx.x;
  const int tid  = threadIdx.x;
  const int lane = tid & 31;
  const int wave = tid >> 5;

  // zero the weight matrices (512 floats, 256 threads)
  ((float*)BxW)[tid] = 0.f;
  ((float*)AyW)[tid] = 0.f;
  __syncthreads();

  if (tid == 0) {
    const float* r = rois + n * 5;
    int b = (int)r[0];
    float x1 = r[1] * SCALE_F, y1 = r[2] * SCALE_F;
    float x2 = r[3] * SCALE_F, y2 = r[4] * SCALE_F;
    float rw = fmaxf(x2 - x1, 1.f), rh = fmaxf(y2 - y1, 1.f);
    float bw = rw * (1.f / PW_N), bh = rh * (1.f / PH_N);
    int   lox[14], loy[14], vx[14], vy[14];
    float frx[14], fry[14];
    int minx = W_IN - 2, miny = H_IN - 2;
    for (int i = 0; i < 14; ++i) {
      float cx = x1 + ((float)i + 0.5f) * bw * 0.5f;
      vx[i] = (cx >= -1.f) && (cx <= (float)W_IN);
      float cc = fminf(fmaxf(cx, 0.f), (float)(W_IN - 1));
      int lo = (int)floorf(cc); if (lo > W_IN - 2) lo = W_IN - 2;
      lox[i] = lo; frx[i] = cc - (float)lo;
      if (vx[i] && lo < minx) minx = lo;

      float cy = y1 + ((float)i + 0.5f) * bh * 0.5f;
      vy[i] = (cy >= -1.f) && (cy <= (float)H_IN);
      float cd = fminf(fmaxf(cy, 0.f), (float)(H_IN - 1));
      int lo2 = (int)floorf(cd); if (lo2 > H_IN - 2) lo2 = H_IN - 2;
      loy[i] = lo2; fry[i] = cd - (float)lo2;
      if (vy[i] && lo2 < miny) miny = lo2;
    }
    int x0 = minx & ~1; if (x0 > W_IN - 16) x0 = W_IN - 16; if (x0 < 0) x0 = 0;
    int y0 = miny & ~1; if (y0 > H_IN - 16) y0 = H_IN - 16; if (y0 < 0) y0 = 0;
    for (int i = 0; i < 14; ++i) {
      int pb = i >> 1; // pooled bin index (SR=2)
      if (vx[i]) {
        int xl = lox[i] - x0; if (xl < 0) xl = 0; if (xl > 14) xl = 14;
        BxW[xl][pb]     += (1.f - frx[i]) * 0.5f;  // 0.5 = 1/SR per axis
        BxW[xl + 1][pb] += frx[i] * 0.5f;
      }
      if (vy[i]) {
        int yl = loy[i] - y0; if (yl < 0) yl = 0; if (yl > 14) yl = 14;
        AyW[pb][yl]     += (1.f - fry[i]) * 0.5f;
        AyW[pb][yl + 1] += fry[i] * 0.5f;
      }
    }
    sp[0] = b; sp[1] = y0; sp[2] = x0;
  }
  __syncthreads();

  const int b  = sp[0], y0 = sp[1], x0 = sp[2];
  const int lm = lane & 15, lh = lane >> 4;

  // Per-ROI constant WMMA fragments (f32 16x16x4 layouts):
  //  A frag: lane = M, VGPR0 holds K = k0+2*lh, VGPR1 holds K = k0+1+2*lh
  //  B frag: lane = N, same K striping
  v2f bx[4], ay[4];
#pragma unroll
  for (int k = 0; k < 4; ++k) {
    int kk = 4 * k + 2 * lh;
    v2f t; t.x = BxW[kk][lm]; t.y = BxW[kk + 1][lm];
    bx[k] = t;
    ay[k] = *(const v2f*)&AyW[lm][kk];   // 8B-aligned (kk even)
  }

  // Global A-fragment base: 16x16 window of the feature map, row = lane%16
  const float* base = data + (size_t)(b * C_TOT) * (H_IN * W_IN)
                           + (size_t)y0 * W_IN + x0;
  const int laneoff = lm * W_IN + 2 * lh;   // even -> 8B-aligned b64 loads

  const int c0 = grp * 128 + wave * 16;
  const float* pbase = base + (size_t)c0 * (H_IN * W_IN);
  float* outb = out + ((size_t)n * C_TOT + c0) * (PH_N * PW_N);

  for (int i = 0; i < 16; ++i) {
    // Stage 1: U = Dwin x Bx   (M=y, K=x, N=pw)
    v8f acc1 = {};
#pragma unroll
    for (int k = 0; k < 4; ++k) {
      v2f a = *(const v2f*)(pbase + laneoff + 4 * k);
      acc1 = __builtin_amdgcn_wmma_f32_16x16x4_f32(
          false, a, false, bx[k], (short)0, acc1, false, false);
    }

    // D-layout -> B-layout via per-wave LDS transpose buffer (U^T[pw][y])
#pragma unroll
    for (int v = 0; v < 8; ++v)
      UT[wave][lm][v + 8 * lh] = acc1[v];
    __builtin_amdgcn_wave_barrier();      // order stores before cross-lane loads

    // Stage 2: out = Ay x U    (M=ph, K=y, N=pw)
    v8f acc2 = {};
#pragma unroll
    for (int k = 0; k < 4; ++k) {
      v2f bu = *(const v2f*)&UT[wave][lm][4 * k + 2 * lh];
      acc2 = __builtin_amdgcn_wmma_f32_16x16x4_f32(
          false, ay[k], false, bu, (short)0, acc2, false, false);
    }
    __builtin_amdgcn_wave_barrier();      // keep next-iter stores after these loads

    // D layout: VGPR v, lanes 0-15 hold out[ph=v][pw=lane]; ph<7, pw<7 used
    if (lane < PW_N) {
#pragma unroll
      for (int v = 0; v < PH_N; ++v)
        outb[v * PW_N + lane] = acc2[v];
    }
    pbase += H_IN * W_IN;
    outb  += PH_N * PW_N;
  }
}

extern "C" void kernel_launch(void* const* d_in, const int* in_sizes, int n_in,
                              void* d_out, int out_size, void* d_ws, size_t ws_size,
                              hipStream_t stream) {
  const float* data = (const float*)d_in[0];   // (2,1024,50,50) f32
  const float* rois = (const float*)d_in[1];   // (512,5) f32
  float* out = (float*)d_out;                  // (512,1024,7,7) f32
  const int nroi = in_sizes[1] / 5;
  dim3 grid(8, (unsigned)nroi);                // 8 channel-groups x 512 ROIs
  roialign_wmma_f32<<<grid, 256, 0, stream>>>(data, rois, out);
}